// Attention_9895604650156
// MI455X (gfx1250) — compile-verified
//
#include <hip/hip_runtime.h>
#include <math.h>

// Sizes fixed by the reference
#define Bsz 32
#define Ssz 2048
#define Hsz 1024

typedef float v2f __attribute__((ext_vector_type(2)));
typedef float v8f __attribute__((ext_vector_type(8)));

// ---------------------------------------------------------------------------
// Kernel 1: qf[b,o] = sum_h query[b,h] * Wq[o,h] + bq[o]   via V_WMMA_F32_16X16X4_F32
// Grid: 128 waves (2 M-tiles of batch x 64 N-tiles of output channel), 1 wave/block.
// A (16x4 f32): lane l holds rows m=l&15, K-pair at kbase=2*(l>>4)  -> float2 from query
// B (4x16 f32): lane l holds col  n=l&15, K-pair at kbase           -> float2 from Wq (B=Wq^T)
// C/D (16x16 f32): VGPR g: lanes 0-15 -> M=g, lanes 16-31 -> M=8+g, N=lane&15
// ---------------------------------------------------------------------------
__global__ __launch_bounds__(32) void qf_wmma_kernel(
    const float* __restrict__ query, const float* __restrict__ Wq,
    const float* __restrict__ bq, float* __restrict__ qf)
{
    const int tile  = blockIdx.x;      // 0..127
    const int mtile = tile & 1;        // 2 tiles of 16 batches
    const int ntile = tile >> 1;       // 64 tiles of 16 output channels
    const int lane  = threadIdx.x;     // 0..31 (wave32)

    const int m = mtile * 16 + (lane & 15);   // batch row for A fragment
    const int n = ntile * 16 + (lane & 15);   // output col for B fragment
    const int khalf = (lane >> 4) * 2;        // 0 or 2

    const float* __restrict__ arow = query + (size_t)m * Hsz;
    const float* __restrict__ brow = Wq    + (size_t)n * Hsz;

    v8f c = {};
    for (int k0 = 0; k0 < Hsz; k0 += 4) {
        const int k = k0 + khalf;
        v2f a = *(const v2f*)(arow + k);
        v2f b = *(const v2f*)(brow + k);
        // 8 args: (neg_a, A, neg_b, B, c_mod, C, reuse_a, reuse_b)
        c = __builtin_amdgcn_wmma_f32_16x16x4_f32(
                false, a, false, b, (short)0, c, false, false);
    }

    const float bias = bq[n];
#pragma unroll
    for (int g = 0; g < 8; ++g) {
        const int row = mtile * 16 + ((lane < 16) ? g : (g + 8));
        qf[(size_t)row * Hsz + n] = c[g] + bias;
    }
}

// ---------------------------------------------------------------------------
// tanh via hardware v_exp_f32: tanh(x) = 1 - 2/(exp(2x)+1)  (saturates correctly)
// ---------------------------------------------------------------------------
__device__ __forceinline__ float fast_tanh(float x) {
    float e = __expf(2.0f * x);
    return 1.0f - 2.0f / (e + 1.0f);
}

// ---------------------------------------------------------------------------
// Kernel 2: align[b,s] = sum_h tanh(qf[b,h] + sf[b,s,h] + cov[b,s]*Wcov[h]) * v[h]
// One wave per (b,s) row; float4 (b128) coalesced loads; streams 256 MB once.
// ---------------------------------------------------------------------------
__global__ __launch_bounds__(256) void align_kernel(
    const float* __restrict__ sf, const float* __restrict__ qf,
    const float* __restrict__ coverage, const float* __restrict__ Wcov,
    const float* __restrict__ vvec, float* __restrict__ align)
{
    const int b    = blockIdx.y;
    const int wave = threadIdx.x >> 5;
    const int lane = threadIdx.x & 31;
    const int s    = blockIdx.x * 8 + wave;

    const float cov = coverage[(size_t)b * Ssz + s];
    const float* __restrict__ row  = sf + ((size_t)b * Ssz + s) * Hsz;
    const float* __restrict__ qrow = qf + (size_t)b * Hsz;

    // Prefetch the stream ~8 rows ahead (speculative; dropped if invalid).
    __builtin_prefetch(row + 8 * Hsz + lane * 16, 0, 1);

    float acc = 0.0f;
#pragma unroll
    for (int it = 0; it < 8; ++it) {
        const int h = it * 128 + lane * 4;
        float4 x  = *(const float4*)(row  + h);
        float4 q4 = *(const float4*)(qrow + h);
        float4 w4 = *(const float4*)(Wcov + h);
        float4 v4 = *(const float4*)(vvec + h);
        acc += fast_tanh(x.x + q4.x + cov * w4.x) * v4.x;
        acc += fast_tanh(x.y + q4.y + cov * w4.y) * v4.y;
        acc += fast_tanh(x.z + q4.z + cov * w4.z) * v4.z;
        acc += fast_tanh(x.w + q4.w + cov * w4.w) * v4.w;
    }
#pragma unroll
    for (int off = 16; off > 0; off >>= 1)
        acc += __shfl_down(acc, off, 32);
    if (lane == 0)
        align[(size_t)b * Ssz + s] = acc;
}

// ---------------------------------------------------------------------------
// Kernel 3: masked softmax over s per batch. One block (256 thr) per batch,
// 8 values per thread held in registers. Writes align_vectors and new_coverage.
// d_out layout: attn_h[32*1024] | new_coverage[32*2048] | align_vectors[32*2048]
// ---------------------------------------------------------------------------
__global__ __launch_bounds__(256) void softmax_kernel(
    const float* __restrict__ align, const unsigned char* __restrict__ mask,
    const float* __restrict__ coverage, float* __restrict__ out)
{
    const int b   = blockIdx.x;
    const int tid = threadIdx.x;
    __shared__ float redmax[8];
    __shared__ float redsum[8];

    float vals[8];
    float m = -INFINITY;
#pragma unroll
    for (int i = 0; i < 8; ++i) {
        const int s = tid + i * 256;
        float x = align[(size_t)b * Ssz + s];
        x = mask[(size_t)b * Ssz + s] ? x : -INFINITY;
        vals[i] = x;
        m = fmaxf(m, x);
    }
#pragma unroll
    for (int off = 16; off > 0; off >>= 1)
        m = fmaxf(m, __shfl_xor(m, off, 32));
    if ((tid & 31) == 0) redmax[tid >> 5] = m;
    __syncthreads();
    float bm = redmax[0];
#pragma unroll
    for (int w = 1; w < 8; ++w) bm = fmaxf(bm, redmax[w]);

    float sum = 0.0f;
#pragma unroll
    for (int i = 0; i < 8; ++i) {
        vals[i] = __expf(vals[i] - bm);
        sum += vals[i];
    }
#pragma unroll
    for (int off = 16; off > 0; off >>= 1)
        sum += __shfl_xor(sum, off, 32);
    if ((tid & 31) == 0) redsum[tid >> 5] = sum;
    __syncthreads();
    float total = 0.0f;
#pragma unroll
    for (int w = 0; w < 8; ++w) total += redsum[w];
    const float inv = 1.0f / total;

    float* __restrict__ ncov = out + Bsz * Hsz;             // new_coverage
    float* __restrict__ avec = out + Bsz * Hsz + Bsz * Ssz; // align_vectors
#pragma unroll
    for (int i = 0; i < 8; ++i) {
        const int s = tid + i * 256;
        const float a = vals[i] * inv;
        const size_t idx = (size_t)b * Ssz + s;
        avec[idx] = a;
        ncov[idx] = coverage[idx] + a;
    }
}

// ---------------------------------------------------------------------------
// Kernel 4: partial[chunk][b][h] = sum_{s in chunk} a[b,s] * states[b,s,h]
// 16 chunks of 128 s each; block covers all 1024 h with float4 per thread.
// Streams the second 256 MB tensor once; deterministic (no atomics).
// ---------------------------------------------------------------------------
__global__ __launch_bounds__(256) void weight_kernel(
    const float* __restrict__ states, const float* __restrict__ avec,
    float* __restrict__ partial)
{
    const int chunk = blockIdx.x;   // 0..15
    const int b     = blockIdx.y;   // 0..31
    const int h     = threadIdx.x * 4;

    const float* __restrict__ base = states + ((size_t)b * Ssz + chunk * 128) * Hsz;
    const float* __restrict__ arow = avec + (size_t)b * Ssz + chunk * 128;

    float4 acc = {0.0f, 0.0f, 0.0f, 0.0f};
    for (int s = 0; s < 128; ++s) {
        __builtin_prefetch(base + (size_t)(s + 8) * Hsz + h, 0, 1);
        const float a = arow[s];
        float4 x = *(const float4*)(base + (size_t)s * Hsz + h);
        acc.x += a * x.x;
        acc.y += a * x.y;
        acc.z += a * x.z;
        acc.w += a * x.w;
    }
    *(float4*)(partial + ((size_t)chunk * Bsz + b) * Hsz + h) = acc;
}

// ---------------------------------------------------------------------------
// Kernel 5: attn_h[b,h] = sum over 16 chunk-partials (final write, no init needed)
// ---------------------------------------------------------------------------
__global__ __launch_bounds__(256) void reduce_kernel(
    const float* __restrict__ partial, float* __restrict__ attn)
{
    const int idx = blockIdx.x * 256 + threadIdx.x;  // 0..32767 == b*1024+h
    float s = 0.0f;
#pragma unroll
    for (int c = 0; c < 16; ++c)
        s += partial[(size_t)c * (Bsz * Hsz) + idx];
    attn[idx] = s;
}

// ---------------------------------------------------------------------------
extern "C" void kernel_launch(void* const* d_in, const int* in_sizes, int n_in,
                              void* d_out, int out_size, void* d_ws, size_t ws_size,
                              hipStream_t stream)
{
    (void)in_sizes; (void)n_in; (void)out_size; (void)ws_size;

    const float*         query    = (const float*)d_in[0];
    const float*         states   = (const float*)d_in[1];
    const float*         sfeat    = (const float*)d_in[2];
    const float*         coverage = (const float*)d_in[3];
    const unsigned char* mask     = (const unsigned char*)d_in[4];
    const float*         Wq       = (const float*)d_in[5];
    const float*         bq       = (const float*)d_in[6];
    const float*         Wcov     = (const float*)d_in[7];
    const float*         vvec     = (const float*)d_in[8];
    float*               out      = (float*)d_out;

    // Workspace layout (floats): qf[32*1024] | align[32*2048] | partial[16*32*1024]
    float* qf      = (float*)d_ws;
    float* align   = qf + Bsz * Hsz;
    float* partial = align + Bsz * Ssz;

    // 1) qf = query @ Wq^T + bq   (WMMA f32 16x16x4)
    qf_wmma_kernel<<<dim3(128), dim3(32), 0, stream>>>(query, Wq, bq, qf);

    // 2) alignment scores (streams states_features, 256 MB)
    align_kernel<<<dim3(Ssz / 8, Bsz), dim3(256), 0, stream>>>(
        sfeat, qf, coverage, Wcov, vvec, align);

    // 3) softmax + align_vectors + new_coverage
    softmax_kernel<<<dim3(Bsz), dim3(256), 0, stream>>>(align, mask, coverage, out);

    // 4) weighted sum partials (streams states, 256 MB)
    weight_kernel<<<dim3(16, Bsz), dim3(256), 0, stream>>>(
        states, out + Bsz * Hsz + Bsz * Ssz, partial);

    // 5) reduce partials -> attn_h
    reduce_kernel<<<dim3((Bsz * Hsz) / 256), dim3(256), 0, stream>>>(partial, out);
}